// Turb2dPDE_19155554140542
// MI455X (gfx1250) — compile-verified
//
#include <hip/hip_runtime.h>
#include <hip/hip_bf16.h>

// CDNA5 / gfx1250, wave32. One workgroup per batch element.
// psi = (1/65^2) * Re( F* . ( S ⊙ (F . x_pad . F) ) . F* ),  F symmetric 65x65 DFT,
// padded to 80x80 so every GEMM is 5x5 tiles of V_WMMA_F32_16X16X4_F32.
// Global->LDS staging uses GLOBAL_LOAD_ASYNC_TO_LDS (ASYNCcnt) where available.

typedef __attribute__((ext_vector_type(2))) float v2f;
typedef __attribute__((ext_vector_type(8))) float v8f;
typedef int vsi4 __attribute__((__vector_size__(4 * sizeof(int))));

#define PT  80            // padded transform size (65 -> 80 = 5*16)
#define MSZ (PT * PT)     // floats per LDS matrix buffer

#if defined(__gfx1250__) && \
    __has_builtin(__builtin_amdgcn_global_load_async_to_lds_b32) && \
    __has_builtin(__builtin_amdgcn_global_load_async_to_lds_b128) && \
    __has_builtin(__builtin_amdgcn_s_wait_asynccnt)
#define HAS_ASYNC_LDS 1
// builtin signatures (from clang diagnostics): b32 takes global int* / lds int*,
// b128 takes global vsi4* / lds vsi4*, then imm offset + imm cpol.
#define G32(p)  ((__attribute__((address_space(1))) int*)(p))
#define L32(p)  ((__attribute__((address_space(3))) int*)(p))
#define G128(p) ((__attribute__((address_space(1))) vsi4*)(p))
#define L128(p) ((__attribute__((address_space(3))) vsi4*)(p))
#else
#define HAS_ASYNC_LDS 0
#endif

// acc += A(80x80) * B(80x80) restricted to output tile (mt,nt), full K sweep.
__device__ __forceinline__ v8f mm_acc(const float* __restrict__ A,
                                      const float* __restrict__ B,
                                      int mt, int nt, int lane, v8f acc) {
  const int r  = mt * 16 + (lane & 15);     // A row (M)
  const int c  = nt * 16 + (lane & 15);     // B col (N)
  const int kh = (lane >> 4) << 1;          // lanes 0-15 -> K 0/1, 16-31 -> K 2/3
#pragma unroll
  for (int kt = 0; kt < PT / 4; ++kt) {
    const int k0 = kt * 4 + kh;
    v2f a, b;
    a.x = A[r * PT + k0];
    a.y = A[r * PT + k0 + 1];
    b.x = B[k0 * PT + c];
    b.y = B[(k0 + 1) * PT + c];
    acc = __builtin_amdgcn_wmma_f32_16x16x4_f32(false, a, false, b,
                                                (short)0, acc, false, false);
  }
  return acc;
}

// Scatter a 16x16 f32 D tile back to LDS (documented C/D layout).
__device__ __forceinline__ void st_tile(float* __restrict__ C, int mt, int nt,
                                        int lane, v8f d) {
  const int r0 = mt * 16 + ((lane >> 4) << 3);   // +8 rows for lanes 16-31
  const int c  = nt * 16 + (lane & 15);
#pragma unroll
  for (int i = 0; i < 8; ++i) C[(r0 + i) * PT + c] = d[i];
}

__global__ void __launch_bounds__(160)
turb2d_rhs_kernel(const float* __restrict__ y0,
                  const long long* __restrict__ env,
                  const float* __restrict__ params,
                  float* __restrict__ out) {
  extern __shared__ float smem[];
  float* FRE = smem;            // Re(F)
  float* FIM = FRE + MSZ;       // Im(F)
  float* B0  = FIM + MSZ;       // W  -> Him/Uim
  float* B1  = B0 + MSZ;        // Gre -> Pre -> x(compact)
  float* B2  = B1 + MSZ;        // Gim -> Pim
  float* B3  = B2 + MSZ;        // Hre/Ure -> psi

  const int b    = blockIdx.x;
  const int tid  = threadIdx.x;
  const int lane = tid & 31;
  const int wv   = tid >> 5;                       // 0..4 : output tile row
  const float* x = y0 + (size_t)b * 4096;

  const int   e    = (int)env[b];
  const float mu   = params[e * 2 + 0];
  const float dom  = (e & 1) ? 1.0f : 0.75f;       // DOMAIN = {.75,1,.75,1,...}
  const float fdx  = dom * (3.14159265358979323846f / 64.0f);
  const float dx2  = fdx * fdx;

  // ---- stage 0: build DFT matrices + wrap-padded input (zero pad to 80) ----
  for (int i = tid; i < MSZ; i += blockDim.x) {
    const int j = i / PT, k = i % PT;
    const bool in65 = (j < 65) & (k < 65);
    float cr = 0.0f, ci = 0.0f;
    if (in65) {
      const float th = 6.2831853071795864769f * (float)((j * k) % 65) * (1.0f / 65.0f);
      cr = __cosf(th);
      ci = -__sinf(th);                            // F = exp(-2*pi*i*jk/65)
    }
    FRE[i] = cr; FIM[i] = ci;
#if HAS_ASYNC_LDS
    if (in65) {
      // per-lane gather: global x (wrapped 64->65) -> LDS w_pad, no VGPR data path
      __builtin_amdgcn_global_load_async_to_lds_b32(
          G32(x + (((j & 63) << 6) | (k & 63))), L32(B0 + i), 0, 0);
    } else {
      B0[i] = 0.0f;
    }
#else
    B0[i] = in65 ? x[((j & 63) << 6) | (k & 63)] : 0.0f;
#endif
  }
#if HAS_ASYNC_LDS
  __builtin_amdgcn_s_wait_asynccnt(0);
#endif
  __syncthreads();

  // ---- stage 2: G = F * W  (W real) ----
  for (int nt = 0; nt < 5; ++nt) {
    v8f gre = {}; v8f gim = {};
    gre = mm_acc(FRE, B0, wv, nt, lane, gre);
    gim = mm_acc(FIM, B0, wv, nt, lane, gim);
    st_tile(B1, wv, nt, lane, gre);
    st_tile(B2, wv, nt, lane, gim);
  }
  __syncthreads();

  // ---- stage 3: H = G * F (complex; F symmetric so F^T = F) ----
  for (int nt = 0; nt < 5; ++nt) {
    v8f t0 = {}; v8f t1 = {};
    t0 = mm_acc(B1, FRE, wv, nt, lane, t0);        // Gre*Fre
    t1 = mm_acc(B2, FIM, wv, nt, lane, t1);        // Gim*Fim
    v8f hre = t0 - t1;
    v8f him = {};
    him = mm_acc(B1, FIM, wv, nt, lane, him);      // Gre*Fim
    him = mm_acc(B2, FRE, wv, nt, lane, him);      // + Gim*Fre
    st_tile(B3, wv, nt, lane, hre);
    st_tile(B0, wv, nt, lane, him);                // W is dead now
  }
  __syncthreads();

  // ---- stage 4: spectral scale in place: U = H * s,  s = 1/(65^2 (eps + k2)) ----
  for (int i = tid; i < MSZ; i += blockDim.x) {
    const int a = i / PT, c = i % PT;
    float s = 0.0f;
    if ((a < 65) & (c < 65) & ((a | c) != 0)) {    // zero DC and pad region
      const int ma = (a <= 32) ? a : a - 65;
      const int mc = (c <= 32) ? c : c - 65;
      const float fa = 0.09666444609f * (float)ma; // 2*pi/65
      const float fc = 0.09666444609f * (float)mc;
      const float k2 = (fa * fa + fc * fc) / dx2;
      s = 1.0f / (4225.0f * (1e-12f + k2));
    }
    B3[i] *= s;
    B0[i] *= s;
  }
  __syncthreads();

  // ---- stage 5: P = conj(F) * U ----
  for (int nt = 0; nt < 5; ++nt) {
    v8f pre = {};
    pre = mm_acc(FRE, B3, wv, nt, lane, pre);      // Fre*Ure
    pre = mm_acc(FIM, B0, wv, nt, lane, pre);      // + Fim*Uim
    v8f t0 = {}; v8f t1 = {};
    t0 = mm_acc(FRE, B0, wv, nt, lane, t0);        // Fre*Uim
    t1 = mm_acc(FIM, B3, wv, nt, lane, t1);        // Fim*Ure
    v8f pim = t0 - t1;
    st_tile(B1, wv, nt, lane, pre);
    st_tile(B2, wv, nt, lane, pim);
  }
  __syncthreads();

  // ---- stage 6: psi = Re(P * conj(F)) = Pre*Fre + Pim*Fim  -> B3 ----
  for (int nt = 0; nt < 5; ++nt) {
    v8f ps = {};
    ps = mm_acc(B1, FRE, wv, nt, lane, ps);
    ps = mm_acc(B2, FIM, wv, nt, lane, ps);
    st_tile(B3, wv, nt, lane, ps);
  }
  __syncthreads();

  // ---- stage 7a: re-stage x compactly into B1 (64x64) ----
#if HAS_ASYNC_LDS
  for (int i = tid * 4; i < 4096; i += blockDim.x * 4) {
    __builtin_amdgcn_global_load_async_to_lds_b128(
        G128(x + i), L128(B1 + i), 0, 0);
  }
  __builtin_amdgcn_s_wait_asynccnt(0);
#else
  for (int i = tid; i < 4096; i += blockDim.x) B1[i] = x[i];
#endif
  __syncthreads();

  // ---- stage 7b: Arakawa Jacobian + Laplacian, write dudt ----
  const float inv4 = 1.0f / (4.0f * dx2);
  const float ilap = 1.0f / dx2;
#define XV(r, c) B1[((r) << 6) | (c)]
#define PS(r, c) B3[(r) * PT + (c)]
  for (int i = tid; i < 4096; i += blockDim.x) {
    const int yy = i >> 6, xx = i & 63;
    const int yp = (yy + 1) & 63, ym = (yy + 63) & 63;
    const int xp = (xx + 1) & 63, xm = (xx + 63) & 63;

    const float x00 = XV(yy, xx);
    const float xpy = XV(yp, xx), xmy = XV(ym, xx);
    const float xpx = XV(yy, xp), xmx = XV(yy, xm);
    const float xpp = XV(yp, xp), xmm = XV(ym, xm);
    const float xpm = XV(yp, xm), xmp = XV(ym, xp);

    const float ppy = PS(yp, xx), pmy = PS(ym, xx);
    const float ppx = PS(yy, xp), pmx = PS(yy, xm);
    const float ppp = PS(yp, xp), pmm = PS(ym, xm);
    const float ppm = PS(yp, xm), pmp = PS(ym, xp);

    const float J1 = (ppy - pmy) * (xpx - xmx) - (ppx - pmx) * (xpy - xmy);
    const float J2 = xpx * (ppp - pmp) - xmx * (ppm - pmm)
                   - xpy * (ppp - ppm) + xmy * (pmp - pmm);
    const float J3 = xpp * (ppy - ppx) - xmm * (pmx - pmy)
                   - xpm * (ppy - pmx) + xmp * (ppx - pmy);
    const float lap = xmy + xpy + xmx + xpx + xpm - 4.0f * x00;

    out[(size_t)b * 4096 + i] =
        -(J1 + J2 + J3) * (inv4 * (1.0f / 3.0f)) + mu * lap * ilap;
  }
#undef XV
#undef PS
}

extern "C" void kernel_launch(void* const* d_in, const int* in_sizes, int n_in,
                              void* d_out, int out_size, void* d_ws, size_t ws_size,
                              hipStream_t stream) {
  (void)n_in; (void)d_ws; (void)ws_size; (void)out_size;
  const float*     y0     = (const float*)d_in[1];      // (B,1,64,64) f32
  const long long* env    = (const long long*)d_in[2];  // (B,) int64
  const float*     params = (const float*)d_in[3];      // (8,2) f32
  float*           out    = (float*)d_out;

  const int B = in_sizes[1] / 4096;                     // 2048
  const size_t shmem = (size_t)(6 * MSZ) * sizeof(float); // 153600 B (< 320KB/WGP)

  (void)hipFuncSetAttribute(reinterpret_cast<const void*>(turb2d_rhs_kernel),
                            hipFuncAttributeMaxDynamicSharedMemorySize, (int)shmem);
  turb2d_rhs_kernel<<<B, 160, shmem, stream>>>(y0, env, params, out);
}